// GINNodeEmbedding_59803124630273
// MI455X (gfx1250) — compile-verified
//
#include <hip/hip_runtime.h>

#define DD 128
#define NL 5
#define NCENT 20

typedef __attribute__((ext_vector_type(2))) float v2f;
typedef __attribute__((ext_vector_type(8))) float v8f;
typedef __attribute__((ext_vector_type(4))) unsigned int v4u;
typedef __attribute__((ext_vector_type(8))) unsigned int v8u;

static __device__ __forceinline__ v8f wmma_f32_4(v2f a, v2f b, v8f c) {
  // D = A(16x4,f32) * B(4x16,f32) + C(16x16,f32)  -> v_wmma_f32_16x16x4_f32
  return __builtin_amdgcn_wmma_f32_16x16x4_f32(false, a, false, b, (short)0, c,
                                               false, false);
}

// LDS byte offset of a generic pointer into __shared__ space.
static __device__ __forceinline__ unsigned int lds_addr_of(const void* p) {
  return (unsigned int)(unsigned long long)(
      __attribute__((address_space(3))) const char*)p;
}

// Issue one flat-1D Tensor Data Mover load: nelem f32 from gsrc -> LDS.
// D# built per CDNA5 ISA 8.3/8.4 (count=1, type=2, data_size=4B,
// tensor = nelem x 1, tile = nelem x 1). VADDR2/3 = NULL (<=2D tensor).
// nelem must be < 65536 (tile_dim0 is 16-bit).
static __device__ __forceinline__ void tdm_load_1d(unsigned int lds_off,
                                                   const float* gsrc,
                                                   unsigned int nelem) {
  unsigned long long ga = (unsigned long long)gsrc;
  v4u g0;
  g0[0] = 1u;                                   // count=1, user mode
  g0[1] = lds_off;                              // lds_addr
  g0[2] = (unsigned int)ga;                     // global_addr[31:0]
  g0[3] = (unsigned int)(ga >> 32) | (2u << 30);  // addr[56:32] | type=2
  v8u g1;
  g1[0] = (2u << 16);            // workgroup_mask=0, data_size=2 (4 bytes)
  g1[1] = (nelem & 0xFFFFu) << 16;            // tensor_dim0[15:0]
  g1[2] = (nelem >> 16) | (1u << 16);         // tensor_dim0[31:16], dim1=1
  g1[3] = (nelem & 0xFFFFu) << 16;            // tile_dim0 = nelem
  g1[4] = 1u;                                  // tile_dim1=1, tile_dim2=0
  g1[5] = nelem;                               // tensor_dim0_stride[31:0]
  g1[6] = 0u;
  g1[7] = 0u;
  asm volatile("tensor_load_to_lds %0, %1" ::"s"(g0), "s"(g1) : "memory");
}

// --------------------------------------------------------------------------
// AtomEncoder: h[n,d] = sum_f atom_emb[x[n,f] + atom_off[f], d]
// float4-vectorized: one thread = one node x 4 channels (global_*_b128).
// --------------------------------------------------------------------------
__global__ void atom_encoder_kernel(const int* __restrict__ x,
                                    const int* __restrict__ atom_off,
                                    const float* __restrict__ atom_emb,
                                    float* __restrict__ h, int N) {
  int idx = blockIdx.x * blockDim.x + threadIdx.x;
  if (idx >= N * (DD / 4)) return;
  int n = idx >> 5;
  int d4 = (idx & 31) << 2;
  float4 acc = make_float4(0.f, 0.f, 0.f, 0.f);
#pragma unroll
  for (int f = 0; f < 9; ++f) {
    int row = x[n * 9 + f] + atom_off[f];
    float4 v = *(const float4*)&atom_emb[row * DD + d4];
    acc.x += v.x;
    acc.y += v.y;
    acc.z += v.z;
    acc.w += v.w;
  }
  *(float4*)&h[n * DD + d4] = acc;
}

// --------------------------------------------------------------------------
// Fused per-layer edge pipeline:
//   e = bond_emb gather-sum + bl@W_bl + b_bl + pr@W_pr + b_pr
//   msg = relu(h[src] + e);  agg[dst] += msg   (atomic scatter)
// One wave handles a 16-edge tile. W_bl/W_pr staged in LDS by the TDM
// (overlapped with the RBF exp computation), then 10 WMMAs per column tile.
// --------------------------------------------------------------------------
__global__ void edge_msg_scatter_kernel(
    const int* __restrict__ src, const int* __restrict__ dst,
    const int* __restrict__ eai, const float* __restrict__ eaf,
    const int* __restrict__ bond_off,
    const float* __restrict__ bemb,   // layer slice [sumB, D]
    const float* __restrict__ Wbl, const float* __restrict__ bbl,
    const float* __restrict__ Wpr, const float* __restrict__ bpr,
    const float* __restrict__ h, float* __restrict__ agg,
    int E, int nTiles) {
  __shared__ float sW[2 * NCENT * DD];  // [0..2559]=W_bl, [2560..]=W_pr
  float* sWbl = sW;
  float* sWpr = sW + NCENT * DD;

  // Wave 0 kicks off both DMA transfers; everyone else proceeds to the
  // (independent) RBF fragment computation while the TDM fills LDS.
  if ((threadIdx.x >> 5) == 0) {
    tdm_load_1d(lds_addr_of(sWbl), Wbl, NCENT * DD);
    tdm_load_1d(lds_addr_of(sWpr), Wpr, NCENT * DD);
  }

  int tile = blockIdx.x * (blockDim.x >> 5) + (threadIdx.x >> 5);
  int lane = threadIdx.x & 31;
  int half = lane >> 4;   // selects K pair (ISA A-matrix 16x4 layout)
  int lm = lane & 15;     // row within tile
  int m0 = tile * 16;

  // RBF A-fragments (computed once, reused for all 8 column tiles)
  int eA = m0 + lm;
  if (eA >= E) eA = E - 1;
  float xbl = eaf[eA * 2 + 0];
  float xpr = eaf[eA * 2 + 1];
  v2f aBL[5], aPR[5];
#pragma unroll
  for (int kk = 0; kk < 5; ++kk) {
    int k0 = kk * 4 + 2 * half;
    float d0 = xbl - 0.1f * (float)k0;
    float d1 = xbl - 0.1f * (float)(k0 + 1);
    aBL[kk].x = __expf(-10.f * d0 * d0);
    aBL[kk].y = __expf(-10.f * d1 * d1);
    float p0 = xpr - 0.05f * (float)k0;
    float p1 = xpr - 0.05f * (float)(k0 + 1);
    aPR[kk].x = __expf(-1.f * p0 * p0);
    aPR[kk].y = __expf(-1.f * p1 * p1);
  }
  int bo0 = bond_off[0], bo1 = bond_off[1], bo2 = bond_off[2];

  if ((threadIdx.x >> 5) == 0) __builtin_amdgcn_s_wait_tensorcnt(0);
  __syncthreads();
  if (tile >= nTiles) return;

  for (int n0 = 0; n0 < DD; n0 += 16) {
    int col = n0 + lm;
    v8f c = {0.f, 0.f, 0.f, 0.f, 0.f, 0.f, 0.f, 0.f};
#pragma unroll
    for (int kk = 0; kk < 5; ++kk) {
      int krow = kk * 4 + 2 * half;
      v2f b;
      b.x = sWbl[krow * DD + col];
      b.y = sWbl[(krow + 1) * DD + col];
      c = wmma_f32_4(aBL[kk], b, c);
      b.x = sWpr[krow * DD + col];
      b.y = sWpr[(krow + 1) * DD + col];
      c = wmma_f32_4(aPR[kk], b, c);
    }
    float bias = bbl[col] + bpr[col];
#pragma unroll
    for (int r = 0; r < 8; ++r) {
      int e = m0 + r + 8 * half;  // ISA C/D layout: VGPR r <-> rows r, r+8
      if (e < E) {
        int i0 = eai[e * 3 + 0] + bo0;
        int i1 = eai[e * 3 + 1] + bo1;
        int i2 = eai[e * 3 + 2] + bo2;
        float v = c[r] + bias + bemb[i0 * DD + col] + bemb[i1 * DD + col] +
                  bemb[i2 * DD + col] + h[src[e] * DD + col];
        v = fmaxf(v, 0.f);
        atomicAdd(&agg[dst[e] * DD + col], v);
      }
    }
  }
}

// --------------------------------------------------------------------------
// Dense 128x128 GEMM over node rows (WMMA f32 16x16x4).
// A (16x128 activation tile) staged in LDS by the block; W (128x128, 64 KB)
// staged in LDS by the TDM, overlapped with the A staging.
// mode 0: A = (1+eps)*h + agg          (GIN combine) -> z1 = A@W1 + b1
// mode 1: A = relu(z1*scale + shift)   (fused BN1)   -> z2 = A@W2 + b2
// Block = 256 threads = 8 waves; wave w owns cols [16w, 16w+16).
// Dynamic LDS: 16*128 (zs) + 128*128 (W) floats = 72 KB.
// --------------------------------------------------------------------------
__global__ void gemm_kernel(const float* __restrict__ in0,
                            const float* __restrict__ in1,
                            const float* __restrict__ eps,
                            const float* __restrict__ ss,
                            const float* __restrict__ W,
                            const float* __restrict__ bias,
                            float* __restrict__ out, int N, int mode) {
  extern __shared__ float smem[];
  float* zs = smem;              // [16 * DD]
  float* sWt = smem + 16 * DD;   // [DD * DD]

  if ((threadIdx.x >> 5) == 0) {
    tdm_load_1d(lds_addr_of(sWt), W, DD * DD);
  }

  int m0 = blockIdx.x * 16;
  if (mode == 0) {
    float s = 1.0f + eps[0];
    for (int i = threadIdx.x; i < 16 * DD; i += blockDim.x) {
      int row = m0 + (i >> 7);
      int ck = i & (DD - 1);
      zs[i] = (row < N) ? (s * in0[row * DD + ck] + in1[row * DD + ck]) : 0.f;
    }
  } else {
    for (int i = threadIdx.x; i < 16 * DD; i += blockDim.x) {
      int row = m0 + (i >> 7);
      int ck = i & (DD - 1);
      zs[i] = (row < N)
                  ? fmaxf(in0[row * DD + ck] * ss[ck] + ss[DD + ck], 0.f)
                  : 0.f;
    }
  }
  if ((threadIdx.x >> 5) == 0) __builtin_amdgcn_s_wait_tensorcnt(0);
  __syncthreads();

  int lane = threadIdx.x & 31;
  int half = lane >> 4;
  int lm = lane & 15;
  int n0 = (threadIdx.x >> 5) * 16;
  int col = n0 + lm;
  v8f c = {0.f, 0.f, 0.f, 0.f, 0.f, 0.f, 0.f, 0.f};
#pragma unroll
  for (int k = 0; k < DD; k += 4) {
    int krow = k + 2 * half;
    v2f a;
    a.x = zs[lm * DD + krow];
    a.y = zs[lm * DD + krow + 1];
    v2f b;
    b.x = sWt[krow * DD + col];
    b.y = sWt[(krow + 1) * DD + col];
    c = wmma_f32_4(a, b, c);
  }
  float bv = bias[col];
#pragma unroll
  for (int r = 0; r < 8; ++r) {
    int row = m0 + r + 8 * half;
    if (row < N) out[row * DD + col] = c[r] + bv;
  }
}

// --------------------------------------------------------------------------
// Batch-norm statistics: per-channel sum & sum-of-squares (atomic reduce).
// --------------------------------------------------------------------------
__global__ void stats_kernel(const float* __restrict__ z,
                             float* __restrict__ stats, int N) {
  int c = threadIdx.x;  // 0..127
  int r0 = blockIdx.x * 128;
  int r1 = r0 + 128;
  if (r1 > N) r1 = N;
  float s = 0.f, s2 = 0.f;
  for (int r = r0; r < r1; ++r) {
    float v = z[r * DD + c];
    s += v;
    s2 += v * v;
  }
  atomicAdd(&stats[c], s);
  atomicAdd(&stats[DD + c], s2);
}

// Fold mean/var/gamma/beta into per-channel scale & shift.
__global__ void bn_finalize_kernel(const float* __restrict__ stats,
                                   const float* __restrict__ g,
                                   const float* __restrict__ b,
                                   float* __restrict__ ss, int N) {
  int c = threadIdx.x;
  float inv = 1.0f / (float)N;
  float mean = stats[c] * inv;
  float var = stats[DD + c] * inv - mean * mean;
  float rs = rsqrtf(var + 1e-5f);
  float sc = g[c] * rs;
  ss[c] = sc;
  ss[DD + c] = b[c] - mean * sc;
}

// float4-vectorized BN apply (+optional ReLU).
__global__ void bn_apply_kernel(const float* __restrict__ z,
                                const float* __restrict__ ss,
                                float* __restrict__ out, int N, int doRelu) {
  int idx = blockIdx.x * blockDim.x + threadIdx.x;
  if (idx >= N * (DD / 4)) return;
  int c4 = (idx & 31) << 2;
  float4 v = *(const float4*)&z[idx * 4];
  float4 sc = *(const float4*)&ss[c4];
  float4 sh = *(const float4*)&ss[DD + c4];
  float4 o;
  o.x = v.x * sc.x + sh.x;
  o.y = v.y * sc.y + sh.y;
  o.z = v.z * sc.z + sh.z;
  o.w = v.w * sc.w + sh.w;
  if (doRelu) {
    o.x = fmaxf(o.x, 0.f);
    o.y = fmaxf(o.y, 0.f);
    o.z = fmaxf(o.z, 0.f);
    o.w = fmaxf(o.w, 0.f);
  }
  *(float4*)&out[idx * 4] = o;
}

// --------------------------------------------------------------------------
extern "C" void kernel_launch(void* const* d_in, const int* in_sizes, int n_in,
                              void* d_out, int out_size, void* d_ws,
                              size_t ws_size, hipStream_t stream) {
  const int* x = (const int*)d_in[0];
  const int* edge_index = (const int*)d_in[1];
  const int* eai = (const int*)d_in[2];
  const float* eaf = (const float*)d_in[3];
  const int* atom_off = (const int*)d_in[4];
  const int* bond_off = (const int*)d_in[5];
  const float* atom_emb = (const float*)d_in[6];
  const float* bond_emb = (const float*)d_in[7];
  const float* W_bl = (const float*)d_in[8];
  const float* b_bl = (const float*)d_in[9];
  const float* W_pr = (const float*)d_in[10];
  const float* b_pr = (const float*)d_in[11];
  const float* mlp_W1 = (const float*)d_in[12];
  const float* mlp_b1 = (const float*)d_in[13];
  const float* bn1_g = (const float*)d_in[14];
  const float* bn1_b = (const float*)d_in[15];
  const float* mlp_W2 = (const float*)d_in[16];
  const float* mlp_b2 = (const float*)d_in[17];
  const float* eps_gin = (const float*)d_in[18];
  const float* bn_g = (const float*)d_in[19];
  const float* bn_b = (const float*)d_in[20];

  int N = in_sizes[0] / 9;
  int E = in_sizes[1] / 2;
  int sumB = in_sizes[7] / (NL * DD);
  size_t ND = (size_t)N * DD;

  float* h = (float*)d_ws;
  float* agg = h + ND;
  float* z1 = agg + ND;
  float* z2 = z1 + ND;
  float* stats = z2 + ND;      // [2*DD]
  float* ss = stats + 2 * DD;  // [2*DD]

  const int* src = edge_index;
  const int* dst = edge_index + E;

  atom_encoder_kernel<<<(N * (DD / 4) + 255) / 256, 256, 0, stream>>>(
      x, atom_off, atom_emb, h, N);

  int nTilesE = (E + 15) / 16;
  int edgeBlocks = (nTilesE + 7) / 8;
  int rowTiles = (N + 15) / 16;
  size_t gemmShmem = (size_t)(16 * DD + DD * DD) * sizeof(float);  // 72 KB

  for (int l = 0; l < NL; ++l) {
    const float* bemb_l = bond_emb + (size_t)l * sumB * DD;
    const float* Wbl_l = W_bl + (size_t)l * NCENT * DD;
    const float* bbl_l = b_bl + (size_t)l * DD;
    const float* Wpr_l = W_pr + (size_t)l * NCENT * DD;
    const float* bpr_l = b_pr + (size_t)l * DD;
    const float* W1_l = mlp_W1 + (size_t)l * DD * DD;
    const float* b1_l = mlp_b1 + (size_t)l * DD;
    const float* g1_l = bn1_g + (size_t)l * DD;
    const float* bb1_l = bn1_b + (size_t)l * DD;
    const float* W2_l = mlp_W2 + (size_t)l * DD * DD;
    const float* b2_l = mlp_b2 + (size_t)l * DD;
    const float* g_l = bn_g + (size_t)l * DD;
    const float* bb_l = bn_b + (size_t)l * DD;

    hipMemsetAsync(agg, 0, ND * sizeof(float), stream);
    edge_msg_scatter_kernel<<<edgeBlocks, 256, 0, stream>>>(
        src, dst, eai, eaf, bond_off, bemb_l, Wbl_l, bbl_l, Wpr_l, bpr_l, h,
        agg, E, nTilesE);

    gemm_kernel<<<rowTiles, 256, gemmShmem, stream>>>(
        h, agg, eps_gin + l, nullptr, W1_l, b1_l, z1, N, 0);

    hipMemsetAsync(stats, 0, 2 * DD * sizeof(float), stream);
    stats_kernel<<<(N + 127) / 128, 128, 0, stream>>>(z1, stats, N);
    bn_finalize_kernel<<<1, DD, 0, stream>>>(stats, g1_l, bb1_l, ss, N);

    gemm_kernel<<<rowTiles, 256, gemmShmem, stream>>>(
        z1, nullptr, nullptr, ss, W2_l, b2_l, z2, N, 1);

    hipMemsetAsync(stats, 0, 2 * DD * sizeof(float), stream);
    stats_kernel<<<(N + 127) / 128, 128, 0, stream>>>(z2, stats, N);
    bn_finalize_kernel<<<1, DD, 0, stream>>>(stats, g_l, bb_l, ss, N);

    float* outp = (l == NL - 1) ? (float*)d_out : h;
    bn_apply_kernel<<<(N * (DD / 4) + 255) / 256, 256, 0, stream>>>(
        z2, ss, outp, N, (l < NL - 1) ? 1 : 0);
  }
  (void)n_in;
  (void)out_size;
  (void)ws_size;
}